// TimestepDecoder_14302241096195
// MI455X (gfx1250) — compile-verified
//
#include <hip/hip_runtime.h>
#include <hip/hip_bf16.h>
#include <math.h>

// ---- problem dims ----
#define B_   32
#define S_   128
#define V_   50000
#define OOV_ 12
#define E_   512
#define H_   1024
#define K2H  2048        // 2*H
#define G3H  3072        // 3*H
#define VS_  (V_ + S_)   // 50128 : score row length (score_g | score_c)
#define OUTW (V_ + OOV_) // 50012 : out row length
#define CHK  128         // K-chunk for async LDS staging

typedef float v2f __attribute__((ext_vector_type(2)));
typedef float v8f __attribute__((ext_vector_type(8)));

__device__ __forceinline__ float sigmf(float x) { return 1.f / (1.f + expf(-x)); }

// ---------------------------------------------------------------------------
// k1: prev = encoder_states[:, -1] @ Ws_w + Ws_b    (32x2048)@(2048x1024)
// ---------------------------------------------------------------------------
__global__ void k_prev(const float* __restrict__ es, const float* __restrict__ Wsw,
                       const float* __restrict__ Wsb, float* __restrict__ prev) {
  int idx = blockIdx.x * blockDim.x + threadIdx.x;   // 32*1024
  int b = idx >> 10, j = idx & (H_ - 1);
  const float* x = es + (size_t)b * S_ * K2H + (size_t)(S_ - 1) * K2H;
  float acc = Wsb[j];
  for (int k = 0; k < K2H; ++k) acc += x[k] * Wsw[(size_t)k * H_ + j];
  prev[idx] = acc;
}

// ---------------------------------------------------------------------------
// k2: GRU cell -> state (written straight into d_out's state slot)
// ---------------------------------------------------------------------------
__global__ void k_gru(const int* __restrict__ inputs, const float* __restrict__ emb,
                      const float* __restrict__ gk, const float* __restrict__ grk,
                      const float* __restrict__ gb, const float* __restrict__ prev,
                      float* __restrict__ state) {
  int idx = blockIdx.x * blockDim.x + threadIdx.x;   // 32*1024
  int b = idx >> 10, j = idx & (H_ - 1);
  const float* e = emb + (size_t)inputs[b] * E_;
  float xz = gb[j], xr = gb[H_ + j], xh = gb[2 * H_ + j];
  for (int k = 0; k < E_; ++k) {
    float ev = e[k];
    const float* r = gk + (size_t)k * G3H;
    xz += ev * r[j]; xr += ev * r[H_ + j]; xh += ev * r[2 * H_ + j];
  }
  float hz = gb[G3H + j], hr = gb[G3H + H_ + j], hh = gb[G3H + 2 * H_ + j];
  const float* pv = prev + (size_t)b * H_;
  for (int k = 0; k < H_; ++k) {
    float p = pv[k];
    const float* r = grk + (size_t)k * G3H;
    hz += p * r[j]; hr += p * r[H_ + j]; hh += p * r[2 * H_ + j];
  }
  float z = sigmf(xz + hz);
  float rg = sigmf(xr + hr);
  float cand = tanhf(xh + rg * hh);
  state[idx] = z * pv[j] + (1.f - z) * cand;
}

// ---------------------------------------------------------------------------
// k3: score_g = state @ Wo_w + Wo_b   (32 x 50000), K=1024
// HBM-bound on Wo_w (200 MB, ~8.8us @ 23.3 TB/s). Wo_w is single-use:
// non-temporal loads keep it from evicting es/Wc_w out of the 192 MB L2.
// Both 16-row M halves share each Wo_w B-fragment -> Wo_w read exactly once.
// ---------------------------------------------------------------------------
__global__ void k_score_g(const float* __restrict__ state, const float* __restrict__ Wow,
                          const float* __restrict__ Wob, float* __restrict__ score) {
  int lane = threadIdx.x & 31;
  int wave = threadIdx.x >> 5;
  int nt = blockIdx.x * (blockDim.x >> 5) + wave;
  if (nt >= V_ / 16) return;                 // uniform per wave
  int n0 = nt * 16;
  int half = lane >> 4, ml = lane & 15;
  v8f acc0 = {0, 0, 0, 0, 0, 0, 0, 0};
  v8f acc1 = {0, 0, 0, 0, 0, 0, 0, 0};
  const float2* st2 = (const float2*)state;
  for (int k = 0; k < H_; k += 4) {
    float2 a0f = st2[((size_t)ml * H_ + k) / 2 + half];          // rows b=0..15
    float2 a1f = st2[((size_t)(16 + ml) * H_ + k) / 2 + half];   // rows b=16..31
    v2f a0 = {a0f.x, a0f.y};
    v2f a1 = {a1f.x, a1f.y};
    v2f bf;
    size_t kr = (size_t)(k + 2 * half);
    bf[0] = __builtin_nontemporal_load(&Wow[kr * V_ + n0 + ml]);
    bf[1] = __builtin_nontemporal_load(&Wow[(kr + 1) * V_ + n0 + ml]);
    acc0 = __builtin_amdgcn_wmma_f32_16x16x4_f32(false, a0, false, bf, (short)0, acc0, false, false);
    acc1 = __builtin_amdgcn_wmma_f32_16x16x4_f32(false, a1, false, bf, (short)0, acc1, false, false);
  }
  float bias = Wob[n0 + ml];
#pragma unroll
  for (int v = 0; v < 8; ++v) {
    score[(size_t)(v + 8 * half) * VS_ + n0 + ml] = acc0[v] + bias;
    score[(size_t)(16 + v + 8 * half) * VS_ + n0 + ml] = acc1[v] + bias;
  }
}

// ---------------------------------------------------------------------------
// k4: fused attention score (17.2 GFLOP fp32, compute-bound):
//   sc[bs,n]    = tanh( encoder_states[bs,:] @ Wc_w[:,n] + Wc_b[n] )
//   score_c[bs] = tanh( sum_n sc[bs,n] * state[b,n] ) + mask
// 32-row M-tile per block (halves Wc_w L2 traffic vs 16: 1 B-load per 2 WMMAs).
// A staged via double-buffered global_load_async_to_lds_b128 (ASYNCcnt):
// chunk c+1 DMA overlaps chunk c WMMAs. sc never touches memory.
// ---------------------------------------------------------------------------
__global__ void __launch_bounds__(256) k_attn_score(const float* __restrict__ es,
    const float* __restrict__ Wcw, const float* __restrict__ Wcb,
    const float* __restrict__ state, const int* __restrict__ enc,
    float* __restrict__ score) {
  __shared__ __align__(16) float lds_a[2][32 * CHK];  // 2 x 16 KB
  __shared__ float lds_part[8][32];
  int tid = threadIdx.x;
  int lane = tid & 31, wave = tid >> 5;
  int half = lane >> 4, ml = lane & 15;
  int bs0 = blockIdx.x * 32;
  int b = bs0 >> 7;                          // 32-row tile never crosses batch row

  v8f acc[2][8];
#pragma unroll
  for (int mt = 0; mt < 2; ++mt)
#pragma unroll
    for (int j = 0; j < 8; ++j) acc[mt][j] = (v8f){0, 0, 0, 0, 0, 0, 0, 0};

  // issue async DMA of one 32xCHK A-chunk into LDS buffer `buf`
  auto issue_chunk = [&](int kc, int buf) {
    for (int i = tid; i < (32 * CHK) / 4; i += 256) {   // 1024 float4, 4/thread
      int row = i >> 5;                 // CHK/4 = 32 float4 per row
      int col = (i & 31) << 2;
      const float* g = es + (size_t)(bs0 + row) * K2H + kc + col;
      unsigned ldsaddr = (unsigned)(uintptr_t)&lds_a[buf][row * CHK + col];
      asm volatile("global_load_async_to_lds_b128 %0, %1, off"
                   :: "v"(ldsaddr), "v"(g) : "memory");
    }
  };

  issue_chunk(0, 0);
  for (int c = 0; c < K2H / CHK; ++c) {               // 16 chunks
    asm volatile("s_wait_asynccnt 0" ::: "memory");   // my chunk-c DMAs done
    __syncthreads();                                  // everyone's done + safe reuse
    if (c + 1 < K2H / CHK) issue_chunk((c + 1) * CHK, (c + 1) & 1);
    const float* buf = lds_a[c & 1];
    const float2* arow0 = (const float2*)&buf[ml * CHK];
    const float2* arow1 = (const float2*)&buf[(16 + ml) * CHK];
#pragma unroll 1
    for (int j = 0; j < 8; ++j) {
      int n0 = (wave * 8 + j) * 16;
      v8f c0 = acc[0][j], c1 = acc[1][j];
      for (int kk = 0; kk < CHK; kk += 4) {
        float2 a0f = arow0[(kk >> 1) + half];
        float2 a1f = arow1[(kk >> 1) + half];
        v2f a0 = {a0f.x, a0f.y};
        v2f a1 = {a1f.x, a1f.y};
        v2f bf;
        size_t kr = (size_t)(c * CHK + kk + 2 * half);
        bf[0] = Wcw[kr * H_ + n0 + ml];
        bf[1] = Wcw[(kr + 1) * H_ + n0 + ml];
        c0 = __builtin_amdgcn_wmma_f32_16x16x4_f32(false, a0, false, bf, (short)0, c0, false, false);
        c1 = __builtin_amdgcn_wmma_f32_16x16x4_f32(false, a1, false, bf, (short)0, c1, false, false);
      }
      acc[0][j] = c0; acc[1][j] = c1;
    }
  }

  // fuse: tanh(+bias), dot with state over this wave's 128 N columns
  float psum[2][8];
#pragma unroll
  for (int mt = 0; mt < 2; ++mt)
#pragma unroll
    for (int v = 0; v < 8; ++v) psum[mt][v] = 0.f;
  for (int j = 0; j < 8; ++j) {
    int n = (wave * 8 + j) * 16 + ml;
    float bias = Wcb[n];
    float stv = state[(size_t)b * H_ + n];
#pragma unroll
    for (int mt = 0; mt < 2; ++mt)
#pragma unroll
      for (int v = 0; v < 8; ++v) psum[mt][v] += tanhf(acc[mt][j][v] + bias) * stv;
  }
  // reduce over the 16 lanes of each half (halves hold disjoint M sets)
#pragma unroll
  for (int mt = 0; mt < 2; ++mt)
#pragma unroll
    for (int v = 0; v < 8; ++v) {
      float p = psum[mt][v];
      for (int off = 1; off < 16; off <<= 1) p += __shfl_xor(p, off, 32);
      psum[mt][v] = p;
    }
  if (ml == 0) {
#pragma unroll
    for (int mt = 0; mt < 2; ++mt)
#pragma unroll
      for (int v = 0; v < 8; ++v)
        lds_part[wave][mt * 16 + v + 8 * half] = psum[mt][v];
  }
  __syncthreads();
  if (tid < 32) {
    int M = tid;
    float tot = 0.f;
    for (int w = 0; w < 8; ++w) tot += lds_part[w][M];
    int s = (bs0 + M) & (S_ - 1);
    float m = (enc[b * S_ + s] == 0) ? -1000.f : 0.f;
    score[(size_t)b * VS_ + V_ + s] = tanhf(tot) + m;
  }
}

// ---------------------------------------------------------------------------
// k5: softmax in place over each 50128-wide score row
// ---------------------------------------------------------------------------
__global__ void k_softmax(float* __restrict__ score) {
  int b = blockIdx.x, tid = threadIdx.x;
  float* row = score + (size_t)b * VS_;
  __shared__ float red[256];
  float mx = -1e30f;
  for (int i = tid; i < VS_; i += 256) mx = fmaxf(mx, row[i]);
  red[tid] = mx; __syncthreads();
  for (int s = 128; s > 0; s >>= 1) { if (tid < s) red[tid] = fmaxf(red[tid], red[tid + s]); __syncthreads(); }
  mx = red[0]; __syncthreads();
  float sum = 0.f;
  for (int i = tid; i < VS_; i += 256) { float e = expf(row[i] - mx); row[i] = e; sum += e; }
  red[tid] = sum; __syncthreads();
  for (int s = 128; s > 0; s >>= 1) { if (tid < s) red[tid] += red[tid + s]; __syncthreads(); }
  float inv = 1.f / red[0]; __syncthreads();
  for (int i = tid; i < VS_; i += 256) row[i] *= inv;
}

// ---------------------------------------------------------------------------
// k6a: out base = [probs[:, :V] | 1e-5 for OOV]  (full overwrite each call)
// ---------------------------------------------------------------------------
__global__ void k_fill(const float* __restrict__ score, float* __restrict__ out) {
  long idx = (long)blockIdx.x * blockDim.x + threadIdx.x;
  if (idx >= (long)B_ * OUTW) return;
  int b = (int)(idx / OUTW), c = (int)(idx % OUTW);
  out[idx] = (c < V_) ? score[(size_t)b * VS_ + c] : 1e-5f;
}

// ---------------------------------------------------------------------------
// k6b: per-batch finalize: copy_mask, scatter-add prob_c into out,
//      weighted = attn @ encoder_states
// ---------------------------------------------------------------------------
__global__ void k_final(const float* __restrict__ score, const int* __restrict__ enc,
                        const int* __restrict__ inputs, const float* __restrict__ es,
                        float* __restrict__ out, float* __restrict__ weighted) {
  int b = blockIdx.x, tid = threadIdx.x;
  __shared__ float attn[S_];
  __shared__ int maskFlag;
  if (tid == 0) maskFlag = 0;
  __syncthreads();
  if (tid < S_ && enc[b * S_ + tid] == inputs[b]) atomicOr(&maskFlag, 1);
  __syncthreads();
  if (tid < S_) {
    float p = score[(size_t)b * VS_ + V_ + tid];
    atomicAdd(&out[(size_t)b * OUTW + enc[b * S_ + tid]], p);   // prob_c_to_g
    attn[tid] = maskFlag ? p : 0.f;
  }
  __syncthreads();
  const float* esb = es + (size_t)b * S_ * K2H;
  for (int h = tid; h < K2H; h += 256) {
    float acc = 0.f;
    for (int s = 0; s < S_; ++s) acc += attn[s] * esb[(size_t)s * K2H + h];
    weighted[(size_t)b * K2H + h] = acc;
  }
}

// ---------------------------------------------------------------------------
extern "C" void kernel_launch(void* const* d_in, const int* in_sizes, int n_in,
                              void* d_out, int out_size, void* d_ws, size_t ws_size,
                              hipStream_t stream) {
  const int*   inputs = (const int*)d_in[0];
  const int*   enc    = (const int*)d_in[1];
  const float* es     = (const float*)d_in[2];
  const float* emb    = (const float*)d_in[3];
  const float* gk     = (const float*)d_in[4];
  const float* grk    = (const float*)d_in[5];
  const float* gb     = (const float*)d_in[6];
  const float* Wsw    = (const float*)d_in[7];
  const float* Wsb    = (const float*)d_in[8];
  const float* Wow    = (const float*)d_in[9];
  const float* Wob    = (const float*)d_in[10];
  const float* Wcw    = (const float*)d_in[11];
  const float* Wcb    = (const float*)d_in[12];

  float* out      = (float*)d_out;                       // (B, V+OOV)
  float* state    = out + (size_t)B_ * OUTW;             // (B, H)
  float* weighted = state + (size_t)B_ * H_;             // (B, 2H)

  float* score = (float*)d_ws;                           // (B, V+S) = 6.4 MB
  float* prev  = score + (size_t)B_ * VS_;               // (B, H)

  k_prev      <<<(B_ * H_) / 256, 256, 0, stream>>>(es, Wsw, Wsb, prev);
  k_gru       <<<(B_ * H_) / 256, 256, 0, stream>>>(inputs, emb, gk, grk, gb, prev, state);
  k_score_g   <<<(V_ / 16 + 3) / 4, 128, 0, stream>>>(state, Wow, Wob, score);
  k_attn_score<<<(B_ * S_) / 32, 256, 0, stream>>>(es, Wcw, Wcb, state, enc, score);
  k_softmax   <<<B_, 256, 0, stream>>>(score);
  k_fill      <<<((long)B_ * OUTW + 255) / 256, 256, 0, stream>>>(score, out);
  k_final     <<<B_, 256, 0, stream>>>(score, enc, inputs, es, out, weighted);
}